// RCNNCell_66778151518253
// MI455X (gfx1250) — compile-verified
//
#include <hip/hip_runtime.h>

// ---------------------------------------------------------------------------
// RK4 diffusion step, fused (k1..k4 in LDS). Laplacian via 10x chained
// V_WMMA_F32_16X16X4_F32 per 16x16 subtile; tile fill via async global->LDS
// (ASYNCcnt) when the builtins are available. Target gfx1250 (CDNA5), wave32.
// ---------------------------------------------------------------------------

typedef __attribute__((ext_vector_type(2))) float v2f;
typedef __attribute__((ext_vector_type(8))) float v8f;
typedef int v4i_vs __attribute__((vector_size(16)));   // matches builtin param

#if defined(__gfx1250__) && \
    __has_builtin(__builtin_amdgcn_global_load_async_to_lds_b128) && \
    __has_builtin(__builtin_amdgcn_global_load_async_to_lds_b32) && \
    __has_builtin(__builtin_amdgcn_s_wait_asynccnt)
#define USE_ASYNC_LDS 1
#else
#define USE_ASYNC_LDS 0
#endif

#define GH 1024
#define GW 1024
#define TILE 64
#define HALO 8
#define BUF 80            // TILE + 2*HALO
#define PITCH 84          // mult of 4 (16B-aligned rows); 84 mod 64 banks = 20,
                          // gcd(20,64)=4 -> 16-lane column walks conflict-free
#define NTHREADS 256
#define CELLS (BUF * BUF)               // 6400
#define PER_THREAD (CELLS / NTHREADS)   // 25

#if USE_ASYNC_LDS
#define AS1 __attribute__((address_space(1)))
#define AS3 __attribute__((address_space(3)))
// Generic LDS pointers carry the LDS byte offset in their low 32 bits
// (ISA 10.2); generic global pointers equal AS1.
__device__ __forceinline__ AS1 int* glb_b32(const void* p) {
    return (AS1 int*)(unsigned long long)p;
}
__device__ __forceinline__ AS3 int* lds_b32(const void* p) {
    return (AS3 int*)(unsigned int)(unsigned long long)p;
}
__device__ __forceinline__ AS1 v4i_vs* glb_b128(const void* p) {
    return (AS1 v4i_vs*)(unsigned long long)p;
}
__device__ __forceinline__ AS3 v4i_vs* lds_b128(const void* p) {
    return (AS3 v4i_vs*)(unsigned int)(unsigned long long)p;
}
#endif

// 1D fourth-order Laplacian taps (unscaled; k/dx^2 applied after the matmul)
__device__ __forceinline__ float coefd(int d) {
    d = d < 0 ? -d : d;
    if (d == 0) return -2.5f;
    if (d == 1) return 4.0f / 3.0f;
    if (d == 2) return -1.0f / 12.0f;
    return 0.0f;
}

__device__ __forceinline__ int clampi(int v, int lo, int hi) {
    return v < lo ? lo : (v > hi ? hi : v);
}

__global__ __launch_bounds__(NTHREADS) void rk4_wave_kernel(
    const float* __restrict__ hin, const float* __restrict__ kparam,
    float* __restrict__ out)
{
    __shared__ __align__(16) float u0[BUF * PITCH];  // padded initial field
    __shared__ __align__(16) float fld[BUF * PITCH]; // current stage field
    __shared__ __align__(16) float lap[BUF * PITCH]; // k_s = (k/dx^2)*L(field)

    const int tid  = threadIdx.x;
    const int img  = blockIdx.z;
    const int row0 = blockIdx.y * TILE - HALO;   // global row of buffer origin
    const int col0 = blockIdx.x * TILE - HALO;

    const float scale = kparam[0] * 10000.0f;    // k / dx^2, dx = 0.01
    const float* __restrict__ hb = hin + (size_t)img * GH * GW;
    float* __restrict__ ob = out + (size_t)img * GH * GW;

    // Fully interior tile: every buffer cell has 1 <= g <= N-2 (no pad/zero).
    const bool interior = (row0 >= 1) && (row0 + BUF - 1 <= GH - 2) &&
                          (col0 >= 1) && (col0 + BUF - 1 <= GW - 2);

    // Pad-source LDS offsets for the stage updates: clamped coord, or -1
    // outside the domain. Identity (= own cell) on interior tiles.
    int srcOff[PER_THREAD];

    // ---- fill u0: zero outside domain, replicate-pad (clamp to [1, N-2]) ----
#if USE_ASYNC_LDS
    if (interior) {
        const float* gbase = hb + (size_t)row0 * GW + col0;
        for (int q = tid; q < CELLS / 4; q += NTHREADS) {   // 16B quads
            int r = q / (BUF / 4);
            int c = (q % (BUF / 4)) * 4;
            __builtin_amdgcn_global_load_async_to_lds_b128(
                glb_b128(gbase + (size_t)r * GW + c),
                lds_b128(&u0[r * PITCH + c]), 0, 0);
        }
        #pragma unroll
        for (int it = 0; it < PER_THREAD; ++it) {
            int idx = tid + it * NTHREADS;
            srcOff[it] = (idx / BUF) * PITCH + (idx % BUF);
        }
    } else {
        for (int it = 0; it < PER_THREAD; ++it) {
            int idx = tid + it * NTHREADS;
            int r = idx / BUF, c = idx % BUF;
            int gr = row0 + r, gc = col0 + c;
            if (gr >= 0 && gr < GH && gc >= 0 && gc < GW) {
                int cr = clampi(gr, 1, GH - 2), cc = clampi(gc, 1, GW - 2);
                srcOff[it] = (r + cr - gr) * PITCH + (c + cc - gc);
                __builtin_amdgcn_global_load_async_to_lds_b32(
                    glb_b32(hb + (size_t)cr * GW + cc),
                    lds_b32(&u0[r * PITCH + c]), 0, 0);
            } else {
                srcOff[it] = -1;
                u0[r * PITCH + c] = 0.0f;
            }
        }
    }
    __builtin_amdgcn_s_wait_asynccnt(0);
#else
    for (int it = 0; it < PER_THREAD; ++it) {
        int idx = tid + it * NTHREADS;
        int r = idx / BUF, c = idx % BUF;
        int gr = row0 + r, gc = col0 + c;
        float v = 0.0f;
        if (gr >= 0 && gr < GH && gc >= 0 && gc < GW) {
            int cr = clampi(gr, 1, GH - 2), cc = clampi(gc, 1, GW - 2);
            srcOff[it] = (r + cr - gr) * PITCH + (c + cc - gc);
            v = hb[(size_t)cr * GW + cc];
        } else {
            srcOff[it] = -1;
        }
        u0[r * PITCH + c] = v;
    }
#endif
    __syncthreads();

    float acc[PER_THREAD];               // k1 + 2k2 + 2k3 + k4 (central cells)
    #pragma unroll
    for (int it = 0; it < PER_THREAD; ++it) acc[it] = 0.0f;

    const int wave = tid >> 5;
    const int lane = tid & 31;
    const int hf   = lane >> 4;          // lane half (wave32)
    const int mn   = lane & 15;          // A/C row (M) or B/C col (N)

    const float astage[3] = {0.25f, 0.25f, 0.5f};  // DT/2, DT/2, DT  (DT = 0.5)

    for (int s = 0; s < 4; ++s) {
        const float* fsrc = (s == 0) ? u0 : fld;   // stage 0 reads u0 directly

        // =========== WMMA pass: lap = scale * Laplacian(fsrc) ==============
        // Per 16x16 subtile: Lap = Sv(16x20) x Fext(20x16) + Fh(16x20) x Sh(20x16)
        // K split into 5 chunks of 4 -> 10x v_wmma_f32_16x16x4_f32 chained.
        for (int st = wave; st < 25; st += 8) {    // wave-uniform (EXEC all 1s)
            const int Rr = (st / 5) * 16;
            const int Cc = (st % 5) * 16;
            v8f d = {0.f, 0.f, 0.f, 0.f, 0.f, 0.f, 0.f, 0.f};

            #pragma unroll
            for (int kc = 0; kc < 5; ++kc) {       // vertical: Sv x Fext
                const int j0 = kc * 4 + hf * 2;    // K index of .x element
                v2f a, b;
                a.x = coefd(j0     - mn - 2);      // Sv[mn][j0]
                a.y = coefd(j0 + 1 - mn - 2);
                int r1 = clampi(Rr - 2 + j0,     0, BUF - 1);
                int r2 = clampi(Rr - 2 + j0 + 1, 0, BUF - 1);
                b.x = fsrc[r1 * PITCH + Cc + mn];  // Fext[j0][mn]
                b.y = fsrc[r2 * PITCH + Cc + mn];
                d = __builtin_amdgcn_wmma_f32_16x16x4_f32(
                        false, a, false, b, (short)0, d, false, false);
            }
            #pragma unroll
            for (int kc = 0; kc < 5; ++kc) {       // horizontal: Fh x Sh
                const int j0 = kc * 4 + hf * 2;
                v2f a, b;
                int c1 = clampi(Cc - 2 + j0,     0, BUF - 1);
                int c2 = clampi(Cc - 2 + j0 + 1, 0, BUF - 1);
                a.x = fsrc[(Rr + mn) * PITCH + c1]; // Fh[mn][j0]
                a.y = fsrc[(Rr + mn) * PITCH + c2];
                b.x = coefd(j0     - mn - 2);       // Sh[j0][mn]
                b.y = coefd(j0 + 1 - mn - 2);
                d = __builtin_amdgcn_wmma_f32_16x16x4_f32(
                        false, a, false, b, (short)0, d, false, false);
            }
            // C/D layout: VGPR i -> row (i + 8*hf), col mn
            #pragma unroll
            for (int i = 0; i < 8; ++i)
                lap[(Rr + i + hf * 8) * PITCH + Cc + mn] = scale * d[i];
        }
        __syncthreads();

        // =========== pointwise pass: accumulate k_s, build next field ======
        const float wgt = (s == 0 || s == 3) ? 1.0f : 2.0f;
        for (int it = 0; it < PER_THREAD; ++it) {
            int idx = tid + it * NTHREADS;
            int r = idx / BUF, c = idx % BUF;
            int off = r * PITCH + c;
            bool central = (unsigned)(r - HALO) < TILE &&
                           (unsigned)(c - HALO) < TILE;
            float lv = lap[off];
            if (central) acc[it] += wgt * lv;

            if (s < 3) {
                // fld = pad_boundary(u0 + a_s * k_s): zero outside domain,
                // border cells replicate the adjacent interior cell.
                int so = srcOff[it];
                float v = 0.0f;
                if (so >= 0) v = u0[so] + astage[s] * lap[so];
                fld[off] = v;
            } else if (central) {
                // u_next = u0 + DT/6 * (k1 + 2k2 + 2k3 + k4), DT/6 = 1/12
                ob[(size_t)(row0 + r) * GW + (col0 + c)] =
                    u0[off] + (0.5f / 6.0f) * acc[it];
            }
        }
        __syncthreads();
    }
}

extern "C" void kernel_launch(void* const* d_in, const int* in_sizes, int n_in,
                              void* d_out, int out_size, void* d_ws, size_t ws_size,
                              hipStream_t stream) {
    const float* h = (const float*)d_in[0];   // [B,1,1024,1024] f32
    const float* k = (const float*)d_in[1];   // [1] f32
    float* out = (float*)d_out;               // [B,1,1024,1024] f32

    const int nimg = in_sizes[0] / (GH * GW);
    dim3 grid(GW / TILE, GH / TILE, nimg);
    rk4_wave_kernel<<<grid, NTHREADS, 0, stream>>>(h, k, out);
}